// GATConvModel_12000138625376
// MI455X (gfx1250) — compile-verified
//
#include <hip/hip_runtime.h>
#include <math.h>

#define NN 100000
#define EE 1600000
#define NG 64
#define CIN 128
#define CH  64
#define HEADS 4
#define HC  256
#define COUT 10
#define BNEPS 1e-5f

typedef float v2f __attribute__((ext_vector_type(2)));
typedef float v8f __attribute__((ext_vector_type(8)));

// ---------- helpers ----------
__device__ __forceinline__ void atomicMaxF(float* addr, float val) {
  // int ordering matches float ordering for >=0; unsigned ordering is reversed for <0
  if (val >= 0.0f) atomicMax((int*)addr, __float_as_int(val));
  else             atomicMin((unsigned int*)addr, __float_as_uint(val));
}
__device__ __forceinline__ float sigmoidf(float x) { return 1.0f / (1.0f + expf(-x)); }

// ---------- generic fill ----------
__global__ void fill_f32(float* p, float v, long n) {
  long i = (long)blockIdx.x * blockDim.x + threadIdx.x;
  if (i < n) p[i] = v;
}

// ---------- mean of edge_attr ----------
__global__ void reduce_sum(const float* __restrict__ x, float* out, int n) {
  float s = 0.0f;
  for (int i = blockIdx.x * blockDim.x + threadIdx.x; i < n; i += gridDim.x * blockDim.x)
    s += x[i];
  for (int o = 16; o > 0; o >>= 1) s += __shfl_down(s, o, 32);
  if ((threadIdx.x & 31) == 0) atomicAdd(out, s);
}
__global__ void scale1(float* p, float s) { p[0] *= s; }

// coef[h] = sum_c lin_e[h*64+c] * att_edge[h*64+c]   (EDGE_DIM==1 collapses a_edge)
__global__ void coef_kernel(const float* __restrict__ lin_e, const float* __restrict__ att_e,
                            float* coef) {
  int h = threadIdx.x;
  if (h < HEADS) {
    float s = 0.0f;
    for (int c = 0; c < CH; ++c) s += lin_e[h * CH + c] * att_e[h * CH + c];
    coef[h] = s;
  }
}

// ---------- fp32 WMMA GEMM: C[N,M] = A[N,K] @ B[K,M] ----------
// Wave computes a 16x64 output stripe: one A fragment is reused by 4 column
// tiles (4 accumulators), K stepped by 4 with V_WMMA_F32_16X16X4_F32.
// Requires: Nrows % 16 == 0, K % 4 == 0, M % 64 == 0.
__global__ void gemm_wmma_f32(const float* __restrict__ A, const float* __restrict__ B,
                              float* __restrict__ C, int Nrows, int K, int M) {
  int lane = threadIdx.x & 31;
  int wv   = threadIdx.x >> 5;
  int tj   = blockIdx.x << 6;              // 64-wide column stripe
  int ti   = ((blockIdx.y << 2) + wv) << 4;
  if (ti >= Nrows) return;                 // uniform per wave
  int half = lane >> 4;                    // 0: lanes 0-15, 1: lanes 16-31
  int l15  = lane & 15;
  int kg   = half << 1;                    // A/B VGPR0 holds K=0 (low lanes) / K=2 (high lanes)
  const float* Arow = A + (size_t)(ti + l15) * K;
  v8f acc[4];
#pragma unroll
  for (int t = 0; t < 4; ++t) acc[t] = (v8f){0.f, 0.f, 0.f, 0.f, 0.f, 0.f, 0.f, 0.f};
  for (int k = 0; k < K; k += 4) {
    v2f a;
    a.x = Arow[k + kg];
    a.y = Arow[k + kg + 1];
    const float* B0 = B + (size_t)(k + kg) * M + tj + l15;
    const float* B1 = B + (size_t)(k + kg + 1) * M + tj + l15;
#pragma unroll
    for (int t = 0; t < 4; ++t) {
      v2f b;
      b.x = B0[t * 16];
      b.y = B1[t * 16];
      acc[t] = __builtin_amdgcn_wmma_f32_16x16x4_f32(false, a, false, b, (short)0, acc[t],
                                                     false, false);
    }
  }
#pragma unroll
  for (int t = 0; t < 4; ++t) {
    int col = tj + t * 16 + l15;
#pragma unroll
    for (int v = 0; v < 8; ++v)
      C[(size_t)(ti + v + (half << 3)) * M + col] = acc[t][v];
  }
}

// ---------- GAT per-node attention scores ----------
__global__ void gat_scores(const float* __restrict__ xs, const float* __restrict__ att_s,
                           const float* __restrict__ att_d, float* a_src, float* a_dst, int N) {
  int idx = blockIdx.x * blockDim.x + threadIdx.x;
  if (idx >= N * HEADS) return;
  int n = idx >> 2, h = idx & 3;
  const float* xp = xs + (size_t)n * HC + h * CH;
  const float* as = att_s + h * CH;
  const float* ad = att_d + h * CH;
  float ss = 0.0f, sd = 0.0f;
  for (int c = 0; c < CH; ++c) { float v = xp[c]; ss += v * as[c]; sd += v * ad[c]; }
  a_src[idx] = ss; a_dst[idx] = sd;
}

// ---------- GAT edge passes (idx in [0, E+N): last N are self loops) ----------
__global__ void gat_edge_max(const int* __restrict__ src, const int* __restrict__ dst,
                             const float* __restrict__ ea, const float* eamean,
                             const float* __restrict__ a_src, const float* __restrict__ a_dst,
                             const float* coef, float* mbuf, int E, int N) {
  int e = blockIdx.x * blockDim.x + threadIdx.x;
  if (e >= E + N) return;
  int s, d; float w;
  if (e < E) { s = src[e]; d = dst[e]; w = ea[e]; } else { s = d = e - E; w = eamean[0]; }
#pragma unroll
  for (int h = 0; h < HEADS; ++h) {
    float a = a_src[s * HEADS + h] + a_dst[d * HEADS + h] + w * coef[h];
    a = a > 0.0f ? a : 0.2f * a;
    atomicMaxF(&mbuf[d * HEADS + h], a);
  }
}

__global__ void gat_edge_exp(const int* __restrict__ src, const int* __restrict__ dst,
                             const float* __restrict__ ea, const float* eamean,
                             const float* __restrict__ a_src, const float* __restrict__ a_dst,
                             const float* coef, const float* __restrict__ mbuf,
                             float* denom, float* exbuf, int E, int N) {
  int e = blockIdx.x * blockDim.x + threadIdx.x;
  if (e >= E + N) return;
  int s, d; float w;
  if (e < E) { s = src[e]; d = dst[e]; w = ea[e]; } else { s = d = e - E; w = eamean[0]; }
#pragma unroll
  for (int h = 0; h < HEADS; ++h) {
    float a = a_src[s * HEADS + h] + a_dst[d * HEADS + h] + w * coef[h];
    a = a > 0.0f ? a : 0.2f * a;
    float ex = expf(a - mbuf[d * HEADS + h]);
    exbuf[(size_t)e * HEADS + h] = ex;
    atomicAdd(&denom[d * HEADS + h], ex);
  }
}

// wave per edge; lane strides 256 channels (8 atomics/lane)
__global__ void gat_edge_agg(const int* __restrict__ src, const int* __restrict__ dst,
                             const float* __restrict__ exbuf, const float* __restrict__ denom,
                             const float* __restrict__ xs, float* agg, int E, int N) {
  int gid  = blockIdx.x * (blockDim.x >> 5) + (threadIdx.x >> 5);
  int lane = threadIdx.x & 31;
  if (gid >= E + N) return;
  int s, d;
  if (gid < E) { s = src[gid]; d = dst[gid]; } else { s = d = gid - E; }
  float al[HEADS];
#pragma unroll
  for (int h = 0; h < HEADS; ++h)
    al[h] = exbuf[(size_t)gid * HEADS + h] / (denom[d * HEADS + h] + 1e-16f);
  const float* xp = xs + (size_t)s * HC;
  float*       op = agg + (size_t)d * HC;
  for (int c = lane; c < HC; c += 32) atomicAdd(&op[c], xp[c] * al[c >> 6]);
}

// x1 = BN1(agg + b) over N*256 (in place allowed)
__global__ void gat1_finalize(const float* __restrict__ agg, const float* __restrict__ bias,
                              const float* g, const float* bb, const float* m, const float* v,
                              float* x1, long n) {
  long idx = (long)blockIdx.x * blockDim.x + threadIdx.x;
  if (idx >= n) return;
  int j = (int)(idx & (HC - 1));
  float val = agg[idx] + bias[j];
  x1[idx] = (val - m[j]) * rsqrtf(v[j] + BNEPS) * g[j] + bb[j];
}

// x2 = BN2(mean_h(agg) + b)
__global__ void gat2_finalize(const float* __restrict__ agg, const float* __restrict__ bias,
                              const float* g, const float* bb, const float* m, const float* v,
                              float* x2, int N) {
  int idx = blockIdx.x * blockDim.x + threadIdx.x;
  if (idx >= N * CH) return;
  int n = idx >> 6, c = idx & 63;
  const float* ap = agg + (size_t)n * HC;
  float val = 0.25f * (ap[c] + ap[64 + c] + ap[128 + c] + ap[192 + c]) + bias[c];
  x2[idx] = (val - m[c]) * rsqrtf(v[c] + BNEPS) * g[c] + bb[c];
}

// ---------- GCN ----------
__global__ void gcn_deg(const int* __restrict__ dst, const float* __restrict__ ea,
                        float* deg, int E) {
  int e = blockIdx.x * blockDim.x + threadIdx.x;
  if (e < E) atomicAdd(&deg[dst[e]], ea[e]);
}
__global__ void gcn_dis(const float* __restrict__ deg, float* dis, int N) {
  int n = blockIdx.x * blockDim.x + threadIdx.x;
  if (n < N) { float dd = deg[n] + 1.0f; dis[n] = dd > 0.0f ? rsqrtf(dd) : 0.0f; }
}
__global__ void gcn_edge_agg(const int* __restrict__ src, const int* __restrict__ dst,
                             const float* __restrict__ ea, const float* __restrict__ dis,
                             const float* __restrict__ xw, float* agg, int E, int N) {
  int gid  = blockIdx.x * (blockDim.x >> 5) + (threadIdx.x >> 5);
  int lane = threadIdx.x & 31;
  if (gid >= E + N) return;
  int s, d; float norm;
  if (gid < E) { s = src[gid]; d = dst[gid]; norm = dis[s] * ea[gid] * dis[d]; }
  else         { s = d = gid - E; norm = dis[s] * dis[s]; }
  const float* xp = xw + (size_t)s * CH;
  float*       op = agg + (size_t)d * CH;
  for (int c = lane; c < CH; c += 32) atomicAdd(&op[c], xp[c] * norm);
}
__global__ void gcn_finalize(const float* __restrict__ agg, const float* __restrict__ bias,
                             const float* g, const float* bb, const float* m, const float* v,
                             const float* __restrict__ x2, float* x3, int N) {
  int idx = blockIdx.x * blockDim.x + threadIdx.x;
  if (idx >= N * CH) return;
  int c = idx & 63;
  float val = agg[idx] + bias[c];
  val = (val - m[c]) * rsqrtf(v[c] + BNEPS) * g[c] + bb[c] + x2[idx];
  x3[idx] = val > 0.0f ? val : 0.01f * val;
}

// ---------- Set2Set ----------
__global__ void s2s_lstm(const float* __restrict__ Wih, const float* __restrict__ Whh,
                         const float* __restrict__ bih, const float* __restrict__ bhh,
                         const float* __restrict__ qstar, float* h, float* c) {
  __shared__ float qs[2 * CH];
  __shared__ float hs[CH];
  __shared__ float gs[4 * CH];
  int b = blockIdx.x, t = threadIdx.x;       // 256 threads
  if (t < 2 * CH) qs[t] = qstar[b * 2 * CH + t];
  if (t < CH)     hs[t] = h[b * CH + t];
  __syncthreads();
  float acc = bih[t] + bhh[t];
  const float* wi = Wih + (size_t)t * (2 * CH);
  for (int k = 0; k < 2 * CH; ++k) acc += qs[k] * wi[k];
  const float* wh = Whh + (size_t)t * CH;
  for (int k = 0; k < CH; ++k) acc += hs[k] * wh[k];
  gs[t] = acc;
  __syncthreads();
  if (t < CH) {
    float i_ = sigmoidf(gs[t]), f_ = sigmoidf(gs[CH + t]);
    float gg = tanhf(gs[2 * CH + t]), o_ = sigmoidf(gs[3 * CH + t]);
    float cn = f_ * c[b * CH + t] + i_ * gg;
    c[b * CH + t] = cn;
    h[b * CH + t] = o_ * tanhf(cn);
  }
}
__global__ void s2s_e(const float* __restrict__ x, const int* __restrict__ batch,
                      const float* __restrict__ h, float* e_node, float* m64, int N) {
  int n = blockIdx.x * blockDim.x + threadIdx.x;
  if (n >= N) return;
  int b = batch[n];
  const float* xp = x + (size_t)n * CH;
  const float* hp = h + b * CH;
  float s = 0.0f;
  for (int c = 0; c < CH; ++c) s += xp[c] * hp[c];
  e_node[n] = s;
  atomicMaxF(&m64[b], s);
}
__global__ void s2s_exden(const int* __restrict__ batch, const float* __restrict__ m64,
                          float* e_node, float* den, int N) {
  int n = blockIdx.x * blockDim.x + threadIdx.x;
  if (n >= N) return;
  int b = batch[n];
  float ex = expf(e_node[n] - m64[b]);
  e_node[n] = ex;
  atomicAdd(&den[b], ex);
}
__global__ void s2s_r(const float* __restrict__ x, const int* __restrict__ batch,
                      const float* __restrict__ e_node, const float* __restrict__ den,
                      float* r, int N) {
  int gid  = blockIdx.x * (blockDim.x >> 5) + (threadIdx.x >> 5);
  int lane = threadIdx.x & 31;
  if (gid >= N) return;
  int b = batch[gid];
  float a = e_node[gid] / (den[b] + 1e-16f);
  const float* xp = x + (size_t)gid * CH;
  for (int c = lane; c < CH; c += 32) atomicAdd(&r[b * CH + c], a * xp[c]);
}
__global__ void s2s_qstar(const float* __restrict__ h, const float* __restrict__ r,
                          float* qstar) {
  int idx = blockIdx.x * blockDim.x + threadIdx.x;
  if (idx >= NG * 2 * CH) return;
  int b = idx >> 7, j = idx & 127;
  qstar[idx] = (j < CH) ? h[b * CH + j] : r[b * CH + j - CH];
}

// ---------- output head ----------
__global__ void lin1_kernel(const float* __restrict__ q, const float* __restrict__ W,
                            const float* __restrict__ bias, float* o1) {
  __shared__ float qs[2 * CH];
  int b = blockIdx.x, t = threadIdx.x;  // 64 threads
  qs[t] = q[b * 128 + t];
  qs[64 + t] = q[b * 128 + 64 + t];
  __syncthreads();
  float s = bias[t];
  for (int k = 0; k < 128; ++k) s += qs[k] * W[k * 64 + t];
  o1[b * 64 + t] = s > 0.0f ? s : 0.01f * s;
}
__global__ void lin2_kernel(const float* __restrict__ o1, const float* __restrict__ W,
                            const float* __restrict__ bias, float* out) {
  __shared__ float os[CH];
  int b = blockIdx.x, t = threadIdx.x;  // 64 threads
  os[t] = o1[b * 64 + t];
  __syncthreads();
  if (t < COUT) {
    float s = bias[t];
    for (int k = 0; k < 64; ++k) s += os[k] * W[k * COUT + t];
    out[b * COUT + t] = s;
  }
}

// ---------- host ----------
static inline dim3 g1(long n, int bs) { return dim3((unsigned)((n + bs - 1) / bs)); }

extern "C" void kernel_launch(void* const* d_in, const int* in_sizes, int n_in,
                              void* d_out, int out_size, void* d_ws, size_t ws_size,
                              hipStream_t stream) {
  const float* x     = (const float*)d_in[0];
  const int*   ei    = (const int*)d_in[1];
  const int*   batch = (const int*)d_in[2];
  const float* ea    = (const float*)d_in[3];
  const int N = in_sizes[0] / CIN;
  const int E = in_sizes[1] / 2;
  const int* src = ei;
  const int* dst = ei + E;

  // params in setup_inputs() dict order
  const float* g1W  = (const float*)d_in[4];
  const float* g1as = (const float*)d_in[5];
  const float* g1ad = (const float*)d_in[6];
  const float* g1le = (const float*)d_in[7];
  const float* g1ae = (const float*)d_in[8];
  const float* g1b  = (const float*)d_in[9];
  const float* bn1g = (const float*)d_in[10];
  const float* bn1b = (const float*)d_in[11];
  const float* bn1m = (const float*)d_in[12];
  const float* bn1v = (const float*)d_in[13];
  const float* g2W  = (const float*)d_in[14];
  const float* g2as = (const float*)d_in[15];
  const float* g2ad = (const float*)d_in[16];
  const float* g2le = (const float*)d_in[17];
  const float* g2ae = (const float*)d_in[18];
  const float* g2b  = (const float*)d_in[19];
  const float* bn2g = (const float*)d_in[20];
  const float* bn2b = (const float*)d_in[21];
  const float* bn2m = (const float*)d_in[22];
  const float* bn2v = (const float*)d_in[23];
  const float* gcW  = (const float*)d_in[24];
  const float* gcb  = (const float*)d_in[25];
  const float* bn3g = (const float*)d_in[26];
  const float* bn3b = (const float*)d_in[27];
  const float* bn3m = (const float*)d_in[28];
  const float* bn3v = (const float*)d_in[29];
  const float* Wih  = (const float*)d_in[30];
  const float* Whh  = (const float*)d_in[31];
  const float* bih  = (const float*)d_in[32];
  const float* bhh  = (const float*)d_in[33];
  const float* l1W  = (const float*)d_in[34];
  const float* l1b  = (const float*)d_in[35];
  const float* l2W  = (const float*)d_in[36];
  const float* l2b  = (const float*)d_in[37];

  float* ws = (float*)d_ws;
  size_t off = 0;
  float* xs     = ws + off; off += (size_t)N * HC;       // xs1 / xs2 / scratch
  float* agg    = ws + off; off += (size_t)N * HC;       // agg1->x1, agg2, agg3
  float* x2     = ws + off; off += (size_t)N * CH;
  float* xw     = ws + off; off += (size_t)N * CH;
  float* x3     = ws + off; off += (size_t)N * CH;
  float* asrc   = ws + off; off += (size_t)N * HEADS;
  float* adst   = ws + off; off += (size_t)N * HEADS;
  float* mbuf   = ws + off; off += (size_t)N * HEADS;
  float* denom  = ws + off; off += (size_t)N * HEADS;
  float* exbuf  = ws + off; off += (size_t)(E + N) * HEADS;
  float* deg    = ws + off; off += (size_t)N;
  float* dis    = ws + off; off += (size_t)N;
  float* eamean = ws + off; off += 1;
  float* coef   = ws + off; off += HEADS;
  float* hbuf   = ws + off; off += NG * CH;
  float* cbuf   = ws + off; off += NG * CH;
  float* qstar  = ws + off; off += NG * 2 * CH;
  float* rbuf   = ws + off; off += NG * CH;
  float* enode  = ws + off; off += (size_t)N;
  float* m64    = ws + off; off += NG;
  float* d64    = ws + off; off += NG;
  float* o1     = ws + off; off += NG * CH;

  const float NEG_INF = -__builtin_huge_valf();
  const long NHC = (long)N * HC;
  const int  ENB = (E + N + 255) / 256;          // thread-per-edge kernels
  const int  EWB = (E + N + 7) / 8;              // wave-per-edge kernels (8 waves/block)
  const dim3 GEMM_GRID_HC(HC / 64, (N / 16 + 3) / 4);
  const dim3 GEMM_GRID_CH(CH / 64, (N / 16 + 3) / 4);

  // edge_attr mean (ea_f for self-loops)
  fill_f32<<<1, 32, 0, stream>>>(eamean, 0.0f, 1);
  reduce_sum<<<256, 256, 0, stream>>>(ea, eamean, E);
  scale1<<<1, 1, 0, stream>>>(eamean, 1.0f / (float)E);

  // ===== GAT1 =====
  coef_kernel<<<1, 32, 0, stream>>>(g1le, g1ae, coef);
  gemm_wmma_f32<<<GEMM_GRID_HC, 128, 0, stream>>>(x, g1W, xs, N, CIN, HC);
  gat_scores<<<g1((long)N * HEADS, 256), 256, 0, stream>>>(xs, g1as, g1ad, asrc, adst, N);
  fill_f32<<<g1((long)N * HEADS, 256), 256, 0, stream>>>(mbuf, NEG_INF, (long)N * HEADS);
  fill_f32<<<g1((long)N * HEADS, 256), 256, 0, stream>>>(denom, 0.0f, (long)N * HEADS);
  fill_f32<<<g1(NHC, 256), 256, 0, stream>>>(agg, 0.0f, NHC);
  gat_edge_max<<<ENB, 256, 0, stream>>>(src, dst, ea, eamean, asrc, adst, coef, mbuf, E, N);
  gat_edge_exp<<<ENB, 256, 0, stream>>>(src, dst, ea, eamean, asrc, adst, coef, mbuf, denom, exbuf, E, N);
  gat_edge_agg<<<EWB, 256, 0, stream>>>(src, dst, exbuf, denom, xs, agg, E, N);
  gat1_finalize<<<g1(NHC, 256), 256, 0, stream>>>(agg, g1b, bn1g, bn1b, bn1m, bn1v, agg, NHC);

  // ===== GAT2 ===== (x1 lives in agg; consumed by GEMM before agg is reused)
  coef_kernel<<<1, 32, 0, stream>>>(g2le, g2ae, coef);
  gemm_wmma_f32<<<GEMM_GRID_HC, 128, 0, stream>>>(agg, g2W, xs, N, HC, HC);
  gat_scores<<<g1((long)N * HEADS, 256), 256, 0, stream>>>(xs, g2as, g2ad, asrc, adst, N);
  fill_f32<<<g1((long)N * HEADS, 256), 256, 0, stream>>>(mbuf, NEG_INF, (long)N * HEADS);
  fill_f32<<<g1((long)N * HEADS, 256), 256, 0, stream>>>(denom, 0.0f, (long)N * HEADS);
  fill_f32<<<g1(NHC, 256), 256, 0, stream>>>(agg, 0.0f, NHC);
  gat_edge_max<<<ENB, 256, 0, stream>>>(src, dst, ea, eamean, asrc, adst, coef, mbuf, E, N);
  gat_edge_exp<<<ENB, 256, 0, stream>>>(src, dst, ea, eamean, asrc, adst, coef, mbuf, denom, exbuf, E, N);
  gat_edge_agg<<<EWB, 256, 0, stream>>>(src, dst, exbuf, denom, xs, agg, E, N);
  gat2_finalize<<<g1((long)N * CH, 256), 256, 0, stream>>>(agg, g2b, bn2g, bn2b, bn2m, bn2v, x2, N);

  // ===== GCN =====
  fill_f32<<<g1(N, 256), 256, 0, stream>>>(deg, 0.0f, N);
  gcn_deg<<<g1(E, 256), 256, 0, stream>>>(dst, ea, deg, E);
  gcn_dis<<<g1(N, 256), 256, 0, stream>>>(deg, dis, N);
  gemm_wmma_f32<<<GEMM_GRID_CH, 128, 0, stream>>>(x2, gcW, xw, N, CH, CH);
  fill_f32<<<g1((long)N * CH, 256), 256, 0, stream>>>(agg, 0.0f, (long)N * CH);
  gcn_edge_agg<<<EWB, 256, 0, stream>>>(src, dst, ea, dis, xw, agg, E, N);
  gcn_finalize<<<g1((long)N * CH, 256), 256, 0, stream>>>(agg, gcb, bn3g, bn3b, bn3m, bn3v, x2, x3, N);

  // ===== Set2Set =====
  fill_f32<<<g1(NG * CH, 256), 256, 0, stream>>>(hbuf, 0.0f, NG * CH);
  fill_f32<<<g1(NG * CH, 256), 256, 0, stream>>>(cbuf, 0.0f, NG * CH);
  fill_f32<<<g1(NG * 2 * CH, 256), 256, 0, stream>>>(qstar, 0.0f, NG * 2 * CH);
  for (int step = 0; step < 10; ++step) {
    s2s_lstm<<<NG, 256, 0, stream>>>(Wih, Whh, bih, bhh, qstar, hbuf, cbuf);
    fill_f32<<<1, 64, 0, stream>>>(m64, NEG_INF, NG);
    fill_f32<<<1, 64, 0, stream>>>(d64, 0.0f, NG);
    fill_f32<<<g1(NG * CH, 256), 256, 0, stream>>>(rbuf, 0.0f, NG * CH);
    s2s_e<<<g1(N, 256), 256, 0, stream>>>(x3, batch, hbuf, enode, m64, N);
    s2s_exden<<<g1(N, 256), 256, 0, stream>>>(batch, m64, enode, d64, N);
    s2s_r<<<g1(N, 8), 256, 0, stream>>>(x3, batch, enode, d64, rbuf, N);
    s2s_qstar<<<g1(NG * 2 * CH, 256), 256, 0, stream>>>(hbuf, rbuf, qstar);
  }

  // ===== head =====
  lin1_kernel<<<NG, 64, 0, stream>>>(qstar, l1W, l1b, o1);
  lin2_kernel<<<NG, 64, 0, stream>>>(o1, l2W, l2b, (float*)d_out);
}